// Block_79061757984916
// MI455X (gfx1250) — compile-verified
//
#include <hip/hip_runtime.h>

typedef __attribute__((ext_vector_type(2))) float v2f;
typedef __attribute__((ext_vector_type(4))) float v4f;
typedef __attribute__((ext_vector_type(8))) float v8f;

#define CONV 64
// LDS row padding: 68 floats (=17 dwords*4). Bank step per row = 68 % 64 = 4,
// so a b64 read by 16 lanes stepping one row each (+ hi-half offset of 2)
// touches 64 distinct banks -> conflict-free WMMA B-fragment loads.
#define BPAD 68

static constexpr int WAVES   = 8;                 // 256 threads / 32 (wave32)
static constexpr int ROWS_W  = 16;                // rows per wave-tile
static constexpr int ROWS_IT = WAVES * ROWS_W;    // 128 rows per iteration
static constexpr int ITERS   = 8;                 // 1024 rows per workgroup

__global__ __launch_bounds__(256)
void block_diag_gemm(const float* __restrict__ x,
                     const float* __restrict__ blocks,
                     float* __restrict__ out,
                     int nRows, int emb)
{
    __shared__ float ldsB[CONV * BPAD];                 // blocks[n], padded rows
    __shared__ float ldsStage[WAVES][ROWS_W * BPAD];    // per-wave D staging

    const int n     = blockIdx.x;          // which diagonal block
    const int tid   = threadIdx.x;
    const int wave  = tid >> 5;
    const int lane  = tid & 31;
    const int laneM = lane & 15;           // M (A) / N (B) index within 16
    const int hi    = lane >> 4;           // 0: K-pair {0,1}, 1: K-pair {2,3}

    // ---- load blocks[n] (64x64 f32) into padded LDS, coalesced b128 ----
    {
        const v4f* src = (const v4f*)(blocks + (size_t)n * CONV * CONV);
        #pragma unroll
        for (int r = 0; r < 4; ++r) {
            int f4  = r * 256 + tid;       // 0..1023 float4s
            int row = f4 >> 4;             // o
            int c4  = f4 & 15;             // c/4
            *(v4f*)&ldsB[row * BPAD + c4 * 4] = src[f4];
        }
    }
    __syncthreads();

    const size_t rowGroupBase = (size_t)blockIdx.y * (ROWS_IT * ITERS);

    for (int it = 0; it < ITERS; ++it) {
        const int rowBase = (int)rowGroupBase + it * ROWS_IT + wave * ROWS_W;
        if (rowBase + ROWS_W > nRows) continue;   // wave-uniform guard

        // ---- A fragments: v_wmma_f32_16x16x4_f32 A layout ----
        // VGPR0: K = 2*hi, VGPR1: K = 2*hi+1 within each K-step of 4.
        const float* aPtr = x + (size_t)(rowBase + laneM) * emb + n * CONV + hi * 2;
        v2f aF[16];
        #pragma unroll
        for (int k = 0; k < 16; ++k)
            aF[k] = *(const v2f*)(aPtr + k * 4);

        if (it + 1 < ITERS)                       // prefetch next row-tile
            __builtin_prefetch(aPtr + (size_t)ROWS_IT * emb, 0, 3);

        // ---- 64x WMMA: 4 o-tiles x 16 K-steps ----
        v8f acc0 = {}, acc1 = {}, acc2 = {}, acc3 = {};
        #pragma unroll
        for (int k = 0; k < 16; ++k) {
            const v2f a   = aF[k];
            const int cOff = k * 4 + hi * 2;      // same K-pair mapping for B
            v2f b0 = *(const v2f*)&ldsB[( 0 + laneM) * BPAD + cOff];
            v2f b1 = *(const v2f*)&ldsB[(16 + laneM) * BPAD + cOff];
            v2f b2 = *(const v2f*)&ldsB[(32 + laneM) * BPAD + cOff];
            v2f b3 = *(const v2f*)&ldsB[(48 + laneM) * BPAD + cOff];
            acc0 = __builtin_amdgcn_wmma_f32_16x16x4_f32(false, a, false, b0,
                                                         (short)0, acc0, false, false);
            acc1 = __builtin_amdgcn_wmma_f32_16x16x4_f32(false, a, false, b1,
                                                         (short)0, acc1, false, false);
            acc2 = __builtin_amdgcn_wmma_f32_16x16x4_f32(false, a, false, b2,
                                                         (short)0, acc2, false, false);
            acc3 = __builtin_amdgcn_wmma_f32_16x16x4_f32(false, a, false, b3,
                                                         (short)0, acc3, false, false);
        }

        // ---- stage D in LDS (padded, conflict-free), then b128 stores ----
        float* st = &ldsStage[wave][0];
        #pragma unroll
        for (int j = 0; j < 8; ++j) {             // VGPR j: M=j (lanes<16), M=j+8
            const int rl = hi * 8 + j;
            st[rl * BPAD +  0 + laneM] = acc0[j];
            st[rl * BPAD + 16 + laneM] = acc1[j];
            st[rl * BPAD + 32 + laneM] = acc2[j];
            st[rl * BPAD + 48 + laneM] = acc3[j];
        }
        // same-wave LDS RAW -> ordered by DScnt (compiler-inserted waits)
        float* oBase = out + (size_t)rowBase * emb + n * CONV;
        #pragma unroll
        for (int i = 0; i < 8; ++i) {
            const int f4  = i * 32 + lane;        // 1024 floats = 256 float4s
            const int row = f4 >> 4;
            const int c4  = f4 & 15;
            v4f v = *(const v4f*)&st[row * BPAD + c4 * 4];
            *(v4f*)(oBase + (size_t)row * emb + c4 * 4) = v;  // 2 rows x 256B / instr
        }
    }
}

extern "C" void kernel_launch(void* const* d_in, const int* in_sizes, int n_in,
                              void* d_out, int out_size, void* d_ws, size_t ws_size,
                              hipStream_t stream) {
    const float* x      = (const float*)d_in[0];
    const float* blocks = (const float*)d_in[1];
    float*       out    = (float*)d_out;

    const int nblk  = in_sizes[1] / (CONV * CONV);   // 64
    const int emb   = nblk * CONV;                   // 4096
    const int nRows = in_sizes[0] / emb;             // 65536

    const int rowsPerWG = ROWS_IT * ITERS;           // 1024
    dim3 grid(nblk, (nRows + rowsPerWG - 1) / rowsPerWG);
    dim3 block(256);
    block_diag_gemm<<<grid, block, 0, stream>>>(x, blocks, out, nRows, emb);
}